// FW_79963701117418
// MI455X (gfx1250) — compile-verified
//
#include <hip/hip_runtime.h>

// Depth-ordered forward warping (z-buffer scatter) for C=3, H=2160, W=3840.
//
// d_out layout (floats): [ out: 3*HW | valid: HW | collision: HW ]
// During the scatter passes we reuse:
//   out region        -> order-encoded u32 per-channel running max (init enc(-inf))
//   valid region      -> z-buffer min depth bits (u32, init +inf bits)
//   collision region  -> u32 hit count (init 0)
// The finalize kernel decodes everything in place. No d_ws needed; the whole
// 166 MB of atomic state fits in the MI455X's 192 MB L2.

namespace {
constexpr int kW  = 3840;
constexpr int kH  = 2160;
constexpr int kHW = kW * kH;            // 8,294,400
constexpr int kN4 = kHW / 4;            // 2,073,600 (HW divisible by 4)
constexpr unsigned kEncNegInf  = 0x007FFFFFu;  // enc(-inf) under sign-flip encoding
constexpr unsigned kPosInfBits = 0x7F800000u;  // bits(+inf)
}

// Monotonic (total-order preserving) float<->u32 encoding so float max can be
// done with GLOBAL_ATOMIC_MAX_U32 (non-returning).
__device__ __forceinline__ unsigned enc_ord(float f) {
  unsigned b = __float_as_uint(f);
  return (b & 0x80000000u) ? ~b : (b | 0x80000000u);
}
__device__ __forceinline__ float dec_ord(unsigned u) {
  unsigned b = (u & 0x80000000u) ? (u & 0x7FFFFFFFu) : ~u;
  return __uint_as_float(b);
}
__device__ __forceinline__ float dec_or_zero(unsigned u) {
  float v = dec_ord(u);
  // reference: jnp.where(isfinite(out), out, 0)  (untouched slots hold -inf)
  return (fabsf(v) <= 3.402823466e+38f) ? v : 0.0f;
}

// Reference target: clip(x+fx, 0, W-1) / clip(y+fy, 0, H-1), truncate to int.
__device__ __forceinline__ int target_idx(int x, int y, float fx, float fy) {
  float tx = fminf(fmaxf((float)x + fx, 0.0f), (float)(kW - 1));
  float ty = fminf(fmaxf((float)y + fy, 0.0f), (float)(kH - 1));
  return (int)ty * kW + (int)tx;   // truncation == floor for clamped >= 0
}

__global__ void fw_init_kernel(unsigned* __restrict__ outenc,
                               unsigned* __restrict__ minb,
                               unsigned* __restrict__ cnt) {
  int t = blockIdx.x * blockDim.x + threadIdx.x;
  if (t >= kN4) return;
  int p = t * 4;
  uint4 vInit = make_uint4(kEncNegInf, kEncNegInf, kEncNegInf, kEncNegInf);
  uint4 vInf  = make_uint4(kPosInfBits, kPosInfBits, kPosInfBits, kPosInfBits);
  uint4 vZero = make_uint4(0u, 0u, 0u, 0u);
  *reinterpret_cast<uint4*>(outenc + p)           = vInit;
  *reinterpret_cast<uint4*>(outenc + kHW + p)     = vInit;
  *reinterpret_cast<uint4*>(outenc + 2 * kHW + p) = vInit;
  *reinterpret_cast<uint4*>(minb + p)             = vInf;
  *reinterpret_cast<uint4*>(cnt + p)              = vZero;
}

// Pass 1: per-source scatter of count (+1) and z-buffer min (depth bits).
// Depth is strictly positive, so u32 compare on raw bits == float compare.
__global__ void fw_pass1_kernel(const float* __restrict__ flowx,
                                const float* __restrict__ flowy,
                                const float* __restrict__ depth,
                                unsigned* __restrict__ minb,
                                unsigned* __restrict__ cnt) {
  int t = blockIdx.x * blockDim.x + threadIdx.x;
  if (t >= kN4) return;
  int p = t * 4;
  // gfx1250 speculative prefetch of the streaming inputs (global_prefetch_b8)
  __builtin_prefetch(flowx + p + 16384, 0, 0);
  __builtin_prefetch(flowy + p + 16384, 0, 0);
  __builtin_prefetch(depth + p + 16384, 0, 0);

  int y  = p / kW;
  int x0 = p - y * kW;  // 4 consecutive pixels stay in one row (W % 4 == 0)

  const float4 fx = *reinterpret_cast<const float4*>(flowx + p);
  const float4 fy = *reinterpret_cast<const float4*>(flowy + p);
  const float4 dd = *reinterpret_cast<const float4*>(depth + p);

  int i0 = target_idx(x0 + 0, y, fx.x, fy.x);
  int i1 = target_idx(x0 + 1, y, fx.y, fy.y);
  int i2 = target_idx(x0 + 2, y, fx.z, fy.z);
  int i3 = target_idx(x0 + 3, y, fx.w, fy.w);

  atomicAdd(&cnt[i0], 1u);  atomicMin(&minb[i0], __float_as_uint(dd.x));
  atomicAdd(&cnt[i1], 1u);  atomicMin(&minb[i1], __float_as_uint(dd.y));
  atomicAdd(&cnt[i2], 1u);  atomicMin(&minb[i2], __float_as_uint(dd.z));
  atomicAdd(&cnt[i3], 1u);  atomicMin(&minb[i3], __float_as_uint(dd.w));
}

// Pass 2: z-test (plain gather of min bits) + encoded per-channel max scatter.
__global__ void fw_pass2_kernel(const float* __restrict__ flowx,
                                const float* __restrict__ flowy,
                                const float* __restrict__ depth,
                                const float* __restrict__ obj,
                                const unsigned* __restrict__ minb,
                                unsigned* __restrict__ outenc) {
  int t = blockIdx.x * blockDim.x + threadIdx.x;
  if (t >= kN4) return;
  int p = t * 4;
  __builtin_prefetch(flowx + p + 16384, 0, 0);
  __builtin_prefetch(flowy + p + 16384, 0, 0);
  __builtin_prefetch(depth + p + 16384, 0, 0);
  __builtin_prefetch(obj + p + 16384, 0, 0);

  int y  = p / kW;
  int x0 = p - y * kW;

  const float4 fx = *reinterpret_cast<const float4*>(flowx + p);
  const float4 fy = *reinterpret_cast<const float4*>(flowy + p);
  const float4 dd = *reinterpret_cast<const float4*>(depth + p);
  const float4 o0 = *reinterpret_cast<const float4*>(obj + p);
  const float4 o1 = *reinterpret_cast<const float4*>(obj + kHW + p);
  const float4 o2 = *reinterpret_cast<const float4*>(obj + 2 * kHW + p);

  int idx[4];
  idx[0] = target_idx(x0 + 0, y, fx.x, fy.x);
  idx[1] = target_idx(x0 + 1, y, fx.y, fy.y);
  idx[2] = target_idx(x0 + 2, y, fx.z, fy.z);
  idx[3] = target_idx(x0 + 3, y, fx.w, fy.w);

  unsigned db[4] = {__float_as_uint(dd.x), __float_as_uint(dd.y),
                    __float_as_uint(dd.z), __float_as_uint(dd.w)};
  float c0[4] = {o0.x, o0.y, o0.z, o0.w};
  float c1[4] = {o1.x, o1.y, o1.z, o1.w};
  float c2[4] = {o2.x, o2.y, o2.z, o2.w};

#pragma unroll
  for (int k = 0; k < 4; ++k) {
    int i = idx[k];
    if (db[k] <= minb[i]) {   // winner == reference's d <= min_d[idx]
      atomicMax(&outenc[i],           enc_ord(c0[k]));
      atomicMax(&outenc[kHW + i],     enc_ord(c1[k]));
      atomicMax(&outenc[2 * kHW + i], enc_ord(c2[k]));
    }
  }
}

// Finalize: decode encoded channel maxima in place; turn count into
// (collision float, valid 0/1) — valid overwrites the min-depth scratch plane.
__global__ void fw_fin_kernel(float* __restrict__ outp,    // 3*HW (aliases outenc)
                              float* __restrict__ validp,  // HW   (aliases minb)
                              float* __restrict__ collp) { // HW   (aliases cnt)
  int t = blockIdx.x * blockDim.x + threadIdx.x;
  if (t >= kN4) return;
  int p = t * 4;

  uint4 c = *reinterpret_cast<const uint4*>(reinterpret_cast<const unsigned*>(collp) + p);
  float4 coll  = make_float4((float)c.x, (float)c.y, (float)c.z, (float)c.w);
  float4 valid = make_float4(c.x ? 1.0f : 0.0f, c.y ? 1.0f : 0.0f,
                             c.z ? 1.0f : 0.0f, c.w ? 1.0f : 0.0f);
  *reinterpret_cast<float4*>(collp + p)  = coll;
  *reinterpret_cast<float4*>(validp + p) = valid;

  const unsigned* oe = reinterpret_cast<const unsigned*>(outp);
#pragma unroll
  for (int ch = 0; ch < 3; ++ch) {
    uint4 u = *reinterpret_cast<const uint4*>(oe + (size_t)ch * kHW + p);
    float4 v = make_float4(dec_or_zero(u.x), dec_or_zero(u.y),
                           dec_or_zero(u.z), dec_or_zero(u.w));
    *reinterpret_cast<float4*>(outp + (size_t)ch * kHW + p) = v;
  }
}

extern "C" void kernel_launch(void* const* d_in, const int* in_sizes, int n_in,
                              void* d_out, int out_size, void* d_ws, size_t ws_size,
                              hipStream_t stream) {
  (void)in_sizes; (void)n_in; (void)out_size; (void)d_ws; (void)ws_size;

  const float* obj   = (const float*)d_in[0];   // (3, H, W)
  const float* flow  = (const float*)d_in[1];   // (2, H, W)
  const float* depth = (const float*)d_in[2];   // (1, H, W)

  float* out = (float*)d_out;
  unsigned* outenc = (unsigned*)out;                         // 3*HW
  unsigned* minb   = (unsigned*)(out + 3 * (size_t)kHW);     // HW (valid plane)
  unsigned* cnt    = (unsigned*)(out + 4 * (size_t)kHW);     // HW (collision plane)

  const float* flowx = flow;
  const float* flowy = flow + kHW;

  dim3 block(256);                       // 8 wave32 waves per workgroup
  dim3 grid((kN4 + 255) / 256);          // 8100 blocks

  fw_init_kernel <<<grid, block, 0, stream>>>(outenc, minb, cnt);
  fw_pass1_kernel<<<grid, block, 0, stream>>>(flowx, flowy, depth, minb, cnt);
  fw_pass2_kernel<<<grid, block, 0, stream>>>(flowx, flowy, depth, obj, minb, outenc);
  fw_fin_kernel  <<<grid, block, 0, stream>>>(out, out + 3 * (size_t)kHW,
                                              out + 4 * (size_t)kHW);
}